// DiscriminativeLoss_2723009265750
// MI455X (gfx1250) — compile-verified
//
#include <hip/hip_runtime.h>
#include <hip/hip_bf16.h>

// Problem constants (match reference)
#define BATCH 8
#define NPTS  262144       // 2^18
#define FDIM  16
#define KCLU  32
#define BK    (BATCH * KCLU)   // 256
#define DELTA_VAR  0.25f
#define DELTA_DIST 1.5f
#define ALPHA 1.0f
#define BETA  1.0f
#define GAMMA 0.001f
#define EPSF  1e-12f

typedef __attribute__((ext_vector_type(2))) float v2f;
typedef __attribute__((ext_vector_type(8))) float v8f;

// ---------------- workspace layout (floats) ----------------
// [0,            BK*FDIM)        cluster sums      (4096)
// [BK*FDIM,     +BK)             cluster counts    (256)
// [.. ,         +BK*FDIM)        cluster means     (4096)
// [.. ,         +BK)             var sums          (256)
#define WS_SUMS   0
#define WS_CNTS   (BK * FDIM)
#define WS_MEANS  (WS_CNTS + BK)
#define WS_VARS   (WS_MEANS + BK * FDIM)
#define WS_TOTAL  (WS_VARS + BK)          // 8704 floats

__global__ void dl_init_kernel(float* ws) {
    int i = blockIdx.x * blockDim.x + threadIdx.x;
    if (i < WS_TOTAL) ws[i] = 0.0f;
}

// Pass 1: segmented sums + counts. All 256 segments live in LDS (17 KB of the
// 320 KB WGP LDS); one global-atomic flush per block.
__global__ void dl_accum_kernel(const float4* __restrict__ emb4,
                                const int* __restrict__ labels,
                                float* __restrict__ sums,
                                float* __restrict__ counts) {
    __shared__ float lsum[BK * FDIM];
    __shared__ float lcnt[BK];
    for (int i = threadIdx.x; i < BK * FDIM; i += blockDim.x) lsum[i] = 0.0f;
    for (int i = threadIdx.x; i < BK; i += blockDim.x) lcnt[i] = 0.0f;
    __syncthreads();

    const long long total = (long long)BATCH * NPTS;
    const long long stride = (long long)gridDim.x * blockDim.x;
    for (long long p = (long long)blockIdx.x * blockDim.x + threadIdx.x;
         p < total; p += stride) {
        int b = (int)(p >> 18);                 // NPTS = 2^18
        int k = labels[p];
        int seg = b * KCLU + k;
        float4 e0 = emb4[p * 4 + 0];
        float4 e1 = emb4[p * 4 + 1];
        float4 e2 = emb4[p * 4 + 2];
        float4 e3 = emb4[p * 4 + 3];
        float* s = &lsum[seg * FDIM];
        atomicAdd(&s[0],  e0.x); atomicAdd(&s[1],  e0.y);
        atomicAdd(&s[2],  e0.z); atomicAdd(&s[3],  e0.w);
        atomicAdd(&s[4],  e1.x); atomicAdd(&s[5],  e1.y);
        atomicAdd(&s[6],  e1.z); atomicAdd(&s[7],  e1.w);
        atomicAdd(&s[8],  e2.x); atomicAdd(&s[9],  e2.y);
        atomicAdd(&s[10], e2.z); atomicAdd(&s[11], e2.w);
        atomicAdd(&s[12], e3.x); atomicAdd(&s[13], e3.y);
        atomicAdd(&s[14], e3.z); atomicAdd(&s[15], e3.w);
        atomicAdd(&lcnt[seg], 1.0f);
    }
    __syncthreads();
    for (int i = threadIdx.x; i < BK * FDIM; i += blockDim.x)
        atomicAdd(&sums[i], lsum[i]);
    for (int i = threadIdx.x; i < BK; i += blockDim.x)
        atomicAdd(&counts[i], lcnt[i]);
}

__global__ void dl_means_kernel(const float* __restrict__ sums,
                                const float* __restrict__ counts,
                                float* __restrict__ means) {
    int i = blockIdx.x * blockDim.x + threadIdx.x;
    if (i < BK * FDIM) {
        float c = fmaxf(counts[i / FDIM], 1.0f);
        means[i] = sums[i] / c;
    }
}

// Pass 2: hinge-variance per point; means cached in LDS (16 KB), embeddings
// likely L2-resident from pass 1 (150 MB working set < 192 MB L2).
__global__ void dl_var_kernel(const float4* __restrict__ emb4,
                              const int* __restrict__ labels,
                              const float* __restrict__ means,
                              float* __restrict__ var_sums) {
    __shared__ float lmean[BK * FDIM];
    __shared__ float lvar[BK];
    for (int i = threadIdx.x; i < BK * FDIM; i += blockDim.x) lmean[i] = means[i];
    for (int i = threadIdx.x; i < BK; i += blockDim.x) lvar[i] = 0.0f;
    __syncthreads();

    const long long total = (long long)BATCH * NPTS;
    const long long stride = (long long)gridDim.x * blockDim.x;
    for (long long p = (long long)blockIdx.x * blockDim.x + threadIdx.x;
         p < total; p += stride) {
        int b = (int)(p >> 18);
        int k = labels[p];
        int seg = b * KCLU + k;
        float4 e0 = emb4[p * 4 + 0];
        float4 e1 = emb4[p * 4 + 1];
        float4 e2 = emb4[p * 4 + 2];
        float4 e3 = emb4[p * 4 + 3];
        const float* m = &lmean[seg * FDIM];
        float d, d2 = 0.0f;
        d = e0.x - m[0];  d2 += d * d;   d = e0.y - m[1];  d2 += d * d;
        d = e0.z - m[2];  d2 += d * d;   d = e0.w - m[3];  d2 += d * d;
        d = e1.x - m[4];  d2 += d * d;   d = e1.y - m[5];  d2 += d * d;
        d = e1.z - m[6];  d2 += d * d;   d = e1.w - m[7];  d2 += d * d;
        d = e2.x - m[8];  d2 += d * d;   d = e2.y - m[9];  d2 += d * d;
        d = e2.z - m[10]; d2 += d * d;   d = e2.w - m[11]; d2 += d * d;
        d = e3.x - m[12]; d2 += d * d;   d = e3.y - m[13]; d2 += d * d;
        d = e3.z - m[14]; d2 += d * d;   d = e3.w - m[15]; d2 += d * d;
        float dist = sqrtf(fmaxf(d2, EPSF));
        float v = fmaxf(dist - DELTA_VAR, 0.0f);
        atomicAdd(&lvar[seg], v * v);
    }
    __syncthreads();
    for (int i = threadIdx.x; i < BK; i += blockDim.x)
        atomicAdd(&var_sums[i], lvar[i]);
}

// Final: one 256-thread block (8 wave32). Wave w computes batch w's 32x32
// Gram matrix G = M*M^T with V_WMMA_F32_16X16X4_F32 (4 tiles x 4 k-steps),
// then pd^2 = |mi|^2 + |mj|^2 - 2*Gij, hinge push-loss, plus var/reg terms.
__global__ void dl_finalize_kernel(const float* __restrict__ means,
                                   const float* __restrict__ counts,
                                   const float* __restrict__ var_sums,
                                   float* __restrict__ out) {
    __shared__ float lmean[BK * FDIM];   // 16 KB: all batch means
    __shared__ float lnrm2[BK];
    __shared__ float red[256];

    int tid = threadIdx.x;               // 0..255 == one segment each
    float nrm2 = 0.0f;
#pragma unroll
    for (int f = 0; f < FDIM; ++f) {
        float m = means[tid * FDIM + f];
        lmean[tid * FDIM + f] = m;
        nrm2 += m * m;
    }
    lnrm2[tid] = nrm2;
    float c = fmaxf(counts[tid], 1.0f);
    float var_part = var_sums[tid] / c;                 // summed over 256 -> /256
    float reg_part = sqrtf(fmaxf(nrm2, EPSF));          // summed over 256 -> /256
    __syncthreads();

    // WMMA phase: no divergence, EXEC all-1s across every wave.
    int wave = tid >> 5;                 // batch index 0..7
    int lane = tid & 31;
    int half = lane >> 4;                // lane group 0-15 / 16-31
    int l15  = lane & 15;
    const float* Mb = &lmean[wave * KCLU * FDIM];       // 32 x 16 row-major

    float hinge = 0.0f;
#pragma unroll
    for (int it = 0; it < 2; ++it) {
#pragma unroll
        for (int jt = 0; jt < 2; ++jt) {
            v8f cacc = {};
#pragma unroll
            for (int kk = 0; kk < FDIM; kk += 4) {
                // A (16x4): lanes 0-15 hold M=lane K=kk+0/kk+1 in v0/v1,
                //           lanes 16-31 hold K=kk+2/kk+3.
                int ra = it * 16 + l15;
                int rb = jt * 16 + l15;
                v2f a, b;
                a.x = Mb[ra * FDIM + kk + half * 2 + 0];
                a.y = Mb[ra * FDIM + kk + half * 2 + 1];
                // B (4x16) = A-rows of the j-tile transposed: B[k][n] = M[jt*16+n][k]
                b.x = Mb[rb * FDIM + kk + half * 2 + 0];
                b.y = Mb[rb * FDIM + kk + half * 2 + 1];
                cacc = __builtin_amdgcn_wmma_f32_16x16x4_f32(
                    /*neg_a=*/false, a, /*neg_b=*/false, b,
                    /*c_mod=*/(short)0, cacc, /*reuse_a=*/false, /*reuse_b=*/false);
            }
            // C/D layout: VGPR v, lanes 0-15: M=v, N=lane; lanes 16-31: M=v+8.
#pragma unroll
            for (int v = 0; v < 8; ++v) {
                int m = it * 16 + v + 8 * half;
                int n = jt * 16 + l15;
                float g  = cacc[v];
                float sq = lnrm2[wave * KCLU + m] + lnrm2[wave * KCLU + n] - 2.0f * g;
                if (m != n) {
                    float pd = sqrtf(fmaxf(sq, 0.0f));
                    float h  = fmaxf(2.0f * DELTA_DIST - pd, 0.0f);
                    hinge += h * h;
                }
            }
        }
    }

    // Combine: var/256, reg/256*GAMMA, hinge / (2*K*(K-1)) / B
    const float hinge_scale = BETA / (2.0f * KCLU * (KCLU - 1) * (float)BATCH);
    red[tid] = var_part * (ALPHA / (float)BK)
             + reg_part * (GAMMA / (float)BK)
             + hinge * hinge_scale;
    __syncthreads();
    for (int s = 128; s > 0; s >>= 1) {
        if (tid < s) red[tid] += red[tid + s];
        __syncthreads();
    }
    if (tid == 0) out[0] = red[0];
}

extern "C" void kernel_launch(void* const* d_in, const int* in_sizes, int n_in,
                              void* d_out, int out_size, void* d_ws, size_t ws_size,
                              hipStream_t stream) {
    const float4* emb4  = (const float4*)d_in[0];   // (B, N, F) fp32, F=16 -> 4x float4
    const int*    labels = (const int*)d_in[1];     // (B, N) integer
    float* ws = (float*)d_ws;
    float* sums   = ws + WS_SUMS;
    float* counts = ws + WS_CNTS;
    float* means  = ws + WS_MEANS;
    float* vars   = ws + WS_VARS;
    float* out    = (float*)d_out;

    dl_init_kernel<<<(WS_TOTAL + 255) / 256, 256, 0, stream>>>(ws);
    dl_accum_kernel<<<1024, 256, 0, stream>>>(emb4, labels, sums, counts);
    dl_means_kernel<<<(BK * FDIM + 255) / 256, 256, 0, stream>>>(sums, counts, means);
    dl_var_kernel<<<1024, 256, 0, stream>>>(emb4, labels, means, vars);
    dl_finalize_kernel<<<1, 256, 0, stream>>>(means, counts, vars, out);
}